// Convs4x44_42752104464462
// MI455X (gfx1250) — compile-verified
//
#include <hip/hip_runtime.h>
#include <stdint.h>

// ---------------------------------------------------------------------------
// Conv(2x2,s2) -> ReLU(-0.2) -> Conv(2x2,s2) -> ReLU(-2.0) -> MLP(4->3->2)
// -> softmax.  B = 1M samples of 8x8 fp32.  Memory-bound (~0.85 FLOP/byte,
// 272 MB @ 23.3 TB/s => ~11.7 us floor): stream each block's 64 KB chunk with
// the CDNA5 Tensor Data Mover into LDS (padded rows => conflict-free b128
// reads), compute in fp32 VALU.  WMMA is deliberately not used: the only
// fp32-preserving shape (16x16x4) would do 16x redundant MACs plus cross-lane
// shuffles and is strictly slower than straight-line v_fma_f32 here.
// ---------------------------------------------------------------------------

#define TPB          256                  // samples (threads) per block
#define SAMPLE_DW    64                   // 8x8 floats per sample
#define PAD_DW       4                    // TDM-inserted LDS padding per row
#define STRIDE_DW    (SAMPLE_DW + PAD_DW) // 68 dwords -> bank-conflict-free
#define CHUNK_DW     (TPB * SAMPLE_DW)    // 16384 dwords = 64 KB per block

typedef __attribute__((ext_vector_type(4))) unsigned int tdm_v4u;
typedef __attribute__((ext_vector_type(8))) int          tdm_v8i;
typedef __attribute__((ext_vector_type(4))) int          tdm_v4i;

__global__ __launch_bounds__(TPB) void convs_mlp_softmax_kernel(
    const float* __restrict__ x,
    const float* __restrict__ c1w, const float* __restrict__ c2w,
    const float* __restrict__ W1,  const float* __restrict__ b1,
    const float* __restrict__ W2,  const float* __restrict__ b2,
    float* __restrict__ out, int B)
{
    __shared__ float lds[TPB * STRIDE_DW];   // 69,632 B (64KB data + pad)

    const int block0 = blockIdx.x * TPB;

    // ---- TDM: DMA this block's 64 KB chunk (global -> LDS), padded rows ----
    if (threadIdx.x < 32) {   // one wave only: TDM ignores EXEC, so branch it
        uint64_t gaddr = (uint64_t)(const void*)(x + (size_t)block0 * SAMPLE_DW);

        tdm_v4u g0;
        g0.x = 1u;                                   // count = 1 (valid D#)
        g0.y = 0u;                                   // lds_addr (sole LDS alloc)
        g0.z = (unsigned)(gaddr & 0xFFFFFFFFu);      // global_addr[31:0]
        g0.w = (unsigned)((gaddr >> 32) & 0x01FFFFFFu) | (2u << 30); // type=2

        tdm_v8i g1;
        g1[0] = (int)((2u << 16)       // data_size = 4 bytes
                    | (1u << 20)       // pad_enable
                    | (5u << 22)       // pad_interval: every 64 dwords
                    | (3u << 25));     // pad_amount: 4 dwords
        g1[1] = (int)((CHUNK_DW & 0xFFFFu) << 16);        // tensor_dim0[15:0]
        g1[2] = (int)(((CHUNK_DW >> 16) & 0xFFFFu)        // tensor_dim0[31:16]
                    | (1u << 16));                        // tensor_dim1 = 1
        g1[3] = (int)((CHUNK_DW & 0xFFFFu) << 16);        // tile_dim0 = 16384
        g1[4] = 0;                                        // tile_dim1/2 unused
        g1[5] = (int)CHUNK_DW;                            // dim0_stride lo
        g1[6] = 0;
        g1[7] = 0;

        tdm_v4i gz = {0, 0, 0, 0};                        // groups 2/3 unused
#if __has_include(<hip/amd_detail/amd_gfx1250_TDM.h>)
        tdm_v8i gz8 = {0, 0, 0, 0, 0, 0, 0, 0};          // clang-23 6-arg form
        __builtin_amdgcn_tensor_load_to_lds(g0, g1, gz, gz, gz8, 0);
#else
        __builtin_amdgcn_tensor_load_to_lds(g0, g1, gz, gz, 0);  // ROCm 7.2
#endif
        __builtin_amdgcn_s_wait_tensorcnt(0);
    }

    // Opaque clobber: clang has no model that tensor_load_to_lds writes LDS,
    // so without this every ds_load from lds[] folds to undef and the whole
    // compute pipeline is dead-code-eliminated (observed: valu=12, ds=0).
    // This zero-instruction asm may "write" lds[] as far as the optimizer
    // knows, forcing the LDS loads and dependent math to materialize.
    asm volatile("" : : "r"(&lds[0]) : "memory");

    __syncthreads();   // tile resident in LDS for the whole block

    // ---- per-thread tiny weights (uniform addresses -> scalar-cached) ----
    const float k00 = c1w[0], k01 = c1w[1], k10 = c1w[2], k11 = c1w[3];
    const float m00 = c2w[0], m01 = c2w[1], m10 = c2w[2], m11 = c2w[3];

    // ---- load my sample from LDS: 16 x b128, stride 68 dw (no conflicts) ----
    const float4* s4 = (const float4*)&lds[threadIdx.x * STRIDE_DW];
    float xv[64];
#pragma unroll
    for (int i = 0; i < 16; ++i) {
        float4 v = s4[i];
        xv[4 * i + 0] = v.x; xv[4 * i + 1] = v.y;
        xv[4 * i + 2] = v.z; xv[4 * i + 3] = v.w;
    }

    // ---- conv1 2x2 s2 (+ shifted relu): 8x8 -> 4x4 ----
    float y1[16];
#pragma unroll
    for (int i = 0; i < 4; ++i) {
#pragma unroll
        for (int j = 0; j < 4; ++j) {
            float a = xv[(2 * i)     * 8 + 2 * j];
            float b = xv[(2 * i)     * 8 + 2 * j + 1];
            float c = xv[(2 * i + 1) * 8 + 2 * j];
            float d = xv[(2 * i + 1) * 8 + 2 * j + 1];
            float v = fmaf(k00, a, fmaf(k01, b, fmaf(k10, c, k11 * d))) - 0.2f;
            y1[i * 4 + j] = fmaxf(v, 0.0f);
        }
    }

    // ---- conv2 2x2 s2 (+ relu(v-2)): 4x4 -> 4 features ----
    float f[4];
#pragma unroll
    for (int p = 0; p < 2; ++p) {
#pragma unroll
        for (int q = 0; q < 2; ++q) {
            float a = y1[(2 * p)     * 4 + 2 * q];
            float b = y1[(2 * p)     * 4 + 2 * q + 1];
            float c = y1[(2 * p + 1) * 4 + 2 * q];
            float d = y1[(2 * p + 1) * 4 + 2 * q + 1];
            float v = fmaf(m00, a, fmaf(m01, b, fmaf(m10, c, m11 * d)));
            f[p * 2 + q] = fmaxf(v - 2.0f, 0.0f);
        }
    }

    // ---- MLP 4 -> 3 -> 2 ----
    float h[3];
#pragma unroll
    for (int j = 0; j < 3; ++j) {
        float v = b1[j];
#pragma unroll
        for (int mI = 0; mI < 4; ++mI) v = fmaf(W1[j * 4 + mI], f[mI], v);
        h[j] = fmaxf(v, 0.0f);
    }
    float r0 = fmaf(W2[0], h[0], fmaf(W2[1], h[1], fmaf(W2[2], h[2], b2[0])));
    float r1 = fmaf(W2[3], h[0], fmaf(W2[4], h[1], fmaf(W2[5], h[2], b2[1])));

    // ---- softmax over 2 classes (v_exp_f32 + v_rcp_f32) ----
    float mx  = fmaxf(r0, r1);
    float e0  = __expf(r0 - mx);
    float e1  = __expf(r1 - mx);
    float inv = __builtin_amdgcn_rcpf(e0 + e1);

    // ---- stores: (classification[B,2], raw[B,2]) concatenated flat ----
    int sidx = block0 + (int)threadIdx.x;
    if (sidx < B) {
        float2* o = (float2*)out;
        o[sidx]     = make_float2(e0 * inv, e1 * inv);  // classification
        o[B + sidx] = make_float2(r0, r1);              // raw logits
    }
}

extern "C" void kernel_launch(void* const* d_in, const int* in_sizes, int n_in,
                              void* d_out, int out_size, void* d_ws, size_t ws_size,
                              hipStream_t stream) {
    (void)n_in; (void)out_size; (void)d_ws; (void)ws_size;
    const float* x   = (const float*)d_in[0];
    const float* c1w = (const float*)d_in[1];
    const float* c2w = (const float*)d_in[2];
    const float* W1  = (const float*)d_in[3];
    const float* b1  = (const float*)d_in[4];
    const float* W2  = (const float*)d_in[5];
    const float* b2  = (const float*)d_in[6];
    float* out = (float*)d_out;

    const int B = in_sizes[0] / 64;          // samples (1,048,576)
    const int blocks = B / TPB;              // 4096 blocks x 256 threads
    convs_mlp_softmax_kernel<<<blocks, TPB, 0, stream>>>(
        x, c1w, c2w, W1, b1, W2, b2, out, B);
}